// SelfProjectionDev_4183298146555
// MI455X (gfx1250) — compile-verified
//
#include <hip/hip_runtime.h>
#include <math.h>

typedef float v2f __attribute__((ext_vector_type(2)));
typedef float v8f __attribute__((ext_vector_type(8)));

namespace {

constexpr int kB = 64, kI = 512, kJ = 512, kP = 32, kD = 8;
constexpr float kEps = 1e-5f;

// shared memory layout (floats)
constexpr int OFF_STG  = 0;       // 8 waves * 512 : per-wave tanh tile staging (16x32)
constexpr int OFF_GP   = 4096;    // 8 * 1024 : per-wave G partials (reused as sbuf later)
constexpr int OFF_PTJ  = 12288;   // 1024 : standardized p_t_xj
constexpr int OFF_PTI  = 13312;   // 1024 : standardized p_t_xi (stored transposed: H[p][q])
constexpr int OFF_MS   = 14336;   // 32 cols * 16 slots * 2 : online softmax (max,sum)
constexpr int OFF_RED  = 15360;   // 256 : generic block reduction
constexpr int OFF_COLP = 15616;   // 32 : column sums of P-matrix
constexpr int OFF_VA   = 15648;   // 32 : o_rel_xj_sum
constexpr int OFF_VB   = 15680;   // 32 : o_rel_xi_sum
constexpr int OFF_FJ   = 15712;   // 32 : fj
constexpr int OFF_FI   = 15744;   // 32 : fi
constexpr int OFF_SC   = 15776;   // 8  : scalars
constexpr int SMEM_FLOATS = 15784; // 63,136 bytes

__device__ __forceinline__ v8f wmma4(v2f a, v2f b, v8f c) {
  // D = A(16x4 f32) * B(4x16 f32) + C(16x16 f32)
  return __builtin_amdgcn_wmma_f32_16x16x4_f32(false, a, false, b, (short)0, c,
                                               false, false);
}

__device__ float block_sum(float v, float* red) {
  const int tid = threadIdx.x;
  red[tid] = v;
  __syncthreads();
  for (int s = 128; s > 0; s >>= 1) {
    if (tid < s) red[tid] += red[tid + s];
    __syncthreads();
  }
  const float r = red[0];
  __syncthreads();
  return r;
}

__device__ float block_max(float v, float* red) {
  const int tid = threadIdx.x;
  red[tid] = v;
  __syncthreads();
  for (int s = 128; s > 0; s >>= 1) {
    if (tid < s) red[tid] = fmaxf(red[tid], red[tid + s]);
    __syncthreads();
  }
  const float r = red[0];
  __syncthreads();
  return r;
}

// One "orientation": rows R (512) of either x (R=i, sum over j) or x^T (R=j, sum over i).
//   Crel = xOr @ Wrel  -> flat softmax -> column sums  -> out_orel[32]
//   T    = tanh(xOr @ Wt); o_t = (T-m)/(s+eps) (never materialized)
//   G    = T^T @ Pm accumulated via WMMA; then
//   out_pt = std(tanh((G - m*colsum(Pm)) / (s+eps)))   (32x32)
template <bool TRANSX>
__device__ void orientation_pass(const float* __restrict__ X,
                                 const float* __restrict__ Wrel,
                                 const float* __restrict__ Wt,
                                 const float* __restrict__ Pm,
                                 float* smem, float* out_pt, float* out_orel) {
  const int tid  = threadIdx.x;
  const int wave = tid >> 5, lane = tid & 31;
  const int h = lane >> 4, lm = lane & 15;
  float* stg   = smem + OFF_STG + wave * 512;
  float* gpart = smem + OFF_GP;
  float* ms    = smem + OFF_MS;
  float* red   = smem + OFF_RED;
  float* colP  = smem + OFF_COLP;
  float* sc    = smem + OFF_SC;

  float cmax0 = -3.0e38f, cmax1 = -3.0e38f, csum0 = 0.f, csum1 = 0.f;
  float tsum = 0.f, tsq = 0.f;
  const v8f vz = {0.f, 0.f, 0.f, 0.f, 0.f, 0.f, 0.f, 0.f};
  v8f g00 = vz, g01 = vz, g10 = vz, g11 = vz;

  for (int itl = 0; itl < 4; ++itl) {
    const int it = wave + itl * 8;        // row-tile 0..31
    const int r  = it * 16 + lm;          // output row for A fragment
    v8f aR0 = vz, aR1 = vz, aT0 = vz, aT1 = vz;

    for (int k0 = 0; k0 < 512; k0 += 4) {
      const int kk = k0 + 2 * h;          // A/B K split across lane halves
      v2f a;
      if (TRANSX) {                       // A = x^T : coalesced column reads
        a.x = X[kk * 512 + r];
        a.y = X[(kk + 1) * 512 + r];
      } else {                            // A = x : 8-byte row reads
        a.x = X[r * 512 + kk];
        a.y = X[r * 512 + kk + 1];
      }
      v2f b;
      b.x = Wrel[kk * 32 + lm];      b.y = Wrel[(kk + 1) * 32 + lm];
      aR0 = wmma4(a, b, aR0);
      b.x = Wrel[kk * 32 + 16 + lm]; b.y = Wrel[(kk + 1) * 32 + 16 + lm];
      aR1 = wmma4(a, b, aR1);
      b.x = Wt[kk * 32 + lm];        b.y = Wt[(kk + 1) * 32 + lm];
      aT0 = wmma4(a, b, aT0);
      b.x = Wt[kk * 32 + 16 + lm];   b.y = Wt[(kk + 1) * 32 + 16 + lm];
      aT1 = wmma4(a, b, aT1);
    }

    // ---- online flat-softmax stats: each lane owns columns lm and 16+lm
#pragma unroll
    for (int e = 0; e < 8; ++e) {
      float v = aR0[e];
      if (v > cmax0) { csum0 *= expf(cmax0 - v); cmax0 = v; }
      csum0 += expf(v - cmax0);
      v = aR1[e];
      if (v > cmax1) { csum1 *= expf(cmax1 - v); cmax1 = v; }
      csum1 += expf(v - cmax1);
    }

    // ---- tanh, moment accumulation, stage 16x32 tile to per-wave LDS
#pragma unroll
    for (int e = 0; e < 8; ++e) {
      const int row = e + 8 * h;          // C/D layout: VGPR e -> row e / e+8
      const float t0 = tanhf(aT0[e]);
      const float t1 = tanhf(aT1[e]);
      tsum += t0 + t1;
      tsq  += t0 * t0 + t1 * t1;
      stg[row * 32 + lm]      = t0;
      stg[row * 32 + 16 + lm] = t1;
    }
    asm volatile("s_wait_dscnt 0" ::: "memory");  // same-wave LDS store->load

    // ---- G += T^T(16K) @ Pm(K,32) : 4 output tiles, K-chunk = this i-tile
#pragma unroll
    for (int kk4 = 0; kk4 < 4; ++kk4) {
      const int kl = kk4 * 4 + 2 * h;
      v2f a0, a1, b0, b1;
      a0.x = stg[kl * 32 + lm];       a0.y = stg[(kl + 1) * 32 + lm];
      a1.x = stg[kl * 32 + 16 + lm];  a1.y = stg[(kl + 1) * 32 + 16 + lm];
      const int ig = it * 16 + kl;
      b0.x = Pm[ig * 32 + lm];        b0.y = Pm[(ig + 1) * 32 + lm];
      b1.x = Pm[ig * 32 + 16 + lm];   b1.y = Pm[(ig + 1) * 32 + 16 + lm];
      g00 = wmma4(a0, b0, g00);
      g01 = wmma4(a0, b1, g01);
      g10 = wmma4(a1, b0, g10);
      g11 = wmma4(a1, b1, g11);
    }
  }

  // write per-wave G partial as a plain [32][32] block
  float* gw = gpart + wave * 1024;
#pragma unroll
  for (int e = 0; e < 8; ++e) {
    const int row = e + 8 * h;
    gw[row * 32 + lm]             = g00[e];
    gw[row * 32 + 16 + lm]        = g01[e];
    gw[(16 + row) * 32 + lm]      = g10[e];
    gw[(16 + row) * 32 + 16 + lm] = g11[e];
  }
  // per-column softmax stats -> slot (wave,half)
  const int slot = wave * 2 + h;
  ms[lm * 32 + slot * 2]            = cmax0;
  ms[lm * 32 + slot * 2 + 1]        = csum0;
  ms[(16 + lm) * 32 + slot * 2]     = cmax1;
  ms[(16 + lm) * 32 + slot * 2 + 1] = csum1;
  __syncthreads();

  // tanh mean / std (ddof=1, N = 512*32)
  const float sum = block_sum(tsum, red);
  const float sq  = block_sum(tsq, red);
  const float N = 16384.f;
  const float m2 = sum / N;
  const float s2 = sqrtf(fmaxf((sq - sum * sum / N) / (N - 1.f), 0.f));

  // column sums of Pm
  {
    const int q = tid & 31, chunk = tid >> 5;
    float ps = 0.f;
    for (int i = chunk * 64; i < chunk * 64 + 64; ++i) ps += Pm[i * 32 + q];
    red[tid] = ps;
    __syncthreads();
    if (tid < 32) {
      float cs = 0.f;
      for (int c = 0; c < 8; ++c) cs += red[c * 32 + tid];
      colP[tid] = cs;
    }
    __syncthreads();
  }

  // merge per-column (max,sum) over 16 slots, then global softmax normalization
  if (tid < 32) {
    float M = -3.0e38f, S = 0.f;
    for (int s = 0; s < 16; ++s) {
      const float m = ms[tid * 32 + s * 2];
      const float v = ms[tid * 32 + s * 2 + 1];
      if (m > M) { S = S * expf(M - m) + v; M = m; }
      else       { S += v * expf(m - M); }
    }
    red[tid * 2] = M;
    red[tid * 2 + 1] = S;
  }
  __syncthreads();
  if (tid == 0) {
    float G = -3.0e38f;
    for (int p = 0; p < 32; ++p) G = fmaxf(G, red[p * 2]);
    sc[0] = G;
  }
  __syncthreads();
  if (tid < 32) ms[tid] = red[tid * 2 + 1] * expf(red[tid * 2] - sc[0]);
  __syncthreads();
  if (tid == 0) {
    float Z = 0.f;
    for (int p = 0; p < 32; ++p) Z += ms[p];
    sc[1] = Z;
  }
  __syncthreads();
  if (tid < 32) out_orel[tid] = ms[tid] / sc[1];
  __syncthreads();

  // finalize p_t = std(tanh((sumG - m2*colP)/(s2+eps)))
  float lt[4];
  float lsum = 0.f, lsq = 0.f;
#pragma unroll
  for (int e = 0; e < 4; ++e) {
    const int idx = tid * 4 + e;
    const int q = idx & 31;
    float gs = 0.f;
    for (int w = 0; w < 8; ++w) gs += gpart[w * 1024 + idx];
    const float raw = (gs - m2 * colP[q]) / (s2 + kEps);
    const float t = tanhf(raw);
    lt[e] = t;
    lsum += t;
    lsq += t * t;
  }
  const float sum3 = block_sum(lsum, red);
  const float sq3  = block_sum(lsq, red);
  const float m3 = sum3 / 1024.f;
  const float s3 = sqrtf(fmaxf((sq3 - sum3 * sum3 / 1024.f) / 1023.f, 0.f));
#pragma unroll
  for (int e = 0; e < 4; ++e) out_pt[tid * 4 + e] = (lt[e] - m3) / (s3 + kEps);
  __syncthreads();
}

}  // namespace

// ---- kernel 0: per-batch mean/std of x (ddof=1), pre-scaled by (I/P)*(J/P)
__global__ __launch_bounds__(256) void sp_stats(const float* __restrict__ x,
                                                float* __restrict__ stats) {
  __shared__ float red[256];
  const int b = blockIdx.x, tid = threadIdx.x;
  const float* xb = x + (size_t)b * (kI * kJ);
  float s = 0.f, sq = 0.f;
  for (int i = tid; i < kI * kJ; i += 256) {
    const float v = xb[i];
    s += v;
    sq += v * v;
  }
  red[tid] = s;
  __syncthreads();
  for (int st = 128; st > 0; st >>= 1) {
    if (tid < st) red[tid] += red[tid + st];
    __syncthreads();
  }
  const float sum = red[0];
  __syncthreads();
  red[tid] = sq;
  __syncthreads();
  for (int st = 128; st > 0; st >>= 1) {
    if (tid < st) red[tid] += red[tid + st];
    __syncthreads();
  }
  if (tid == 0) {
    const float N = (float)(kI * kJ);
    const float mean = sum / N;
    const float var = (red[0] - sum * sum / N) / (N - 1.f);
    const float scale = ((float)kI / kP) * ((float)kJ / kP);  // 256
    stats[b * 2 + 0] = mean * scale;
    stats[b * 2 + 1] = sqrtf(fmaxf(var, 0.f)) * scale;
  }
}

// ---- kernel 1: one workgroup per (b,d); full fused pipeline
__global__ __launch_bounds__(256, 1) void sp_main(
    const float* __restrict__ x,
    const float* __restrict__ o_xj, const float* __restrict__ o_xi,
    const float* __restrict__ p_xj, const float* __restrict__ p_xi,
    const float* __restrict__ rel_o_xj, const float* __restrict__ rel_o_xi,
    const float* __restrict__ rel_p_xj, const float* __restrict__ rel_p_xi,
    const float* __restrict__ gamma, const float* __restrict__ beta,
    const float* __restrict__ stats, float* __restrict__ contrib) {
  __shared__ float smem[SMEM_FLOATS];
  const int b = blockIdx.x / kD;
  const int d = blockIdx.x % kD;
  const int tid = threadIdx.x;
  const float* X = x + (size_t)b * kI * kJ;
  const size_t wd = (size_t)d * 512 * 32;

  orientation_pass<false>(X, rel_o_xj + wd, o_xj + wd, p_xj + wd, smem,
                          smem + OFF_PTJ, smem + OFF_VA);
  orientation_pass<true>(X, rel_o_xi + wd, o_xi + wd, p_xi + wd, smem,
                         smem + OFF_PTI, smem + OFF_VB);

  float* ptj  = smem + OFF_PTJ;   // p_t_xj[p][q]
  float* pti  = smem + OFF_PTI;   // H[p][q] == p_t_xi[q][p]
  float* sbuf = smem + OFF_GP;    // reuse G-partial region
  float* red  = smem + OFF_RED;
  float* sc   = smem + OFF_SC;
  const float* Rpj = rel_p_xj + (size_t)d * 1024;
  const float* Rpi = rel_p_xi + (size_t)d * 1024;

  // fj[p] = sqrt(orel_j[p] / flatsoftmax(ptj@Rpj).colsum[p])
  {
    float lmax = -3.0e38f;
#pragma unroll
    for (int e = 0; e < 4; ++e) {
      const int idx = tid * 4 + e;
      const int a = idx >> 5, r = idx & 31;
      float s = 0.f;
      for (int c = 0; c < 32; ++c) s += ptj[a * 32 + c] * Rpj[c * 32 + r];
      sbuf[idx] = s;
      lmax = fmaxf(lmax, s);
    }
    __syncthreads();
    const float gmax = block_max(lmax, red);
    if (tid < 32) {
      float cs = 0.f;
      for (int a = 0; a < 32; ++a) cs += expf(sbuf[a * 32 + tid] - gmax);
      red[tid] = cs;
    }
    __syncthreads();
    if (tid == 0) {
      float Z = 0.f;
      for (int p = 0; p < 32; ++p) Z += red[p];
      sc[2] = Z;
    }
    __syncthreads();
    if (tid < 32) {
      const float prel = red[tid] / sc[2];
      smem[OFF_FJ + tid] = sqrtf(smem[OFF_VA + tid] / prel);
    }
    __syncthreads();
  }
  // fi[q] = sqrt(orel_i[q] / flatsoftmax(p_t_xi@Rpi).colsum[q]); p_t_xi[a][c]=H[c][a]
  {
    float lmax = -3.0e38f;
#pragma unroll
    for (int e = 0; e < 4; ++e) {
      const int idx = tid * 4 + e;
      const int a = idx >> 5, r = idx & 31;
      float s = 0.f;
      for (int c = 0; c < 32; ++c) s += pti[c * 32 + a] * Rpi[c * 32 + r];
      sbuf[idx] = s;
      lmax = fmaxf(lmax, s);
    }
    __syncthreads();
    const float gmax = block_max(lmax, red);
    if (tid < 32) {
      float cs = 0.f;
      for (int a = 0; a < 32; ++a) cs += expf(sbuf[a * 32 + tid] - gmax);
      red[tid] = cs;
    }
    __syncthreads();
    if (tid == 0) {
      float Z = 0.f;
      for (int p = 0; p < 32; ++p) Z += red[p];
      sc[2] = Z;
    }
    __syncthreads();
    if (tid < 32) {
      const float prel = red[tid] / sc[2];
      smem[OFF_FI + tid] = sqrtf(smem[OFF_VB + tid] / prel);
    }
    __syncthreads();
  }
  // xb[p][q] = tanh(ptj*fj[p]*H*fi[q]); std; affine; write contribution
  {
    float lv[4];
    float lsum = 0.f, lsq = 0.f;
#pragma unroll
    for (int e = 0; e < 4; ++e) {
      const int idx = tid * 4 + e;
      const int p = idx >> 5, q = idx & 31;
      const float v =
          tanhf(ptj[idx] * smem[OFF_FJ + p] * pti[idx] * smem[OFF_FI + q]);
      lv[e] = v;
      lsum += v;
      lsq += v * v;
    }
    const float sum4 = block_sum(lsum, red);
    const float sq4  = block_sum(lsq, red);
    const float m4 = sum4 / 1024.f;
    const float s4 =
        sqrtf(fmaxf((sq4 - sum4 * sum4 / 1024.f) / 1023.f, 0.f));
    const float xm = stats[b * 2 + 0];
    const float xs = stats[b * 2 + 1];
    const float gm = gamma[d], bt = beta[d];
    float* cw = contrib + (size_t)blockIdx.x * 1024;
#pragma unroll
    for (int e = 0; e < 4; ++e) {
      const int idx = tid * 4 + e;
      const float v = ((lv[e] - m4) / (s4 + kEps)) * xs + xm;
      cw[idx] = v * gm + bt;
    }
  }
}

// ---- kernel 2: deterministic reduce over d
__global__ void sp_finish(const float* __restrict__ contrib,
                          float* __restrict__ out) {
  const int idx = blockIdx.x * 256 + threadIdx.x;
  if (idx >= kB * kP * kP) return;
  const int b = idx >> 10, pq = idx & 1023;
  float s = 0.f;
  for (int d = 0; d < kD; ++d) s += contrib[((size_t)b * kD + d) * 1024 + pq];
  out[idx] = s * (1.f / kD);
}

extern "C" void kernel_launch(void* const* d_in, const int* in_sizes, int n_in,
                              void* d_out, int out_size, void* d_ws,
                              size_t ws_size, hipStream_t stream) {
  (void)in_sizes; (void)n_in; (void)out_size; (void)ws_size;
  const float* x        = (const float*)d_in[0];
  const float* o_xj     = (const float*)d_in[1];
  const float* o_xi     = (const float*)d_in[2];
  const float* p_xj     = (const float*)d_in[3];
  const float* p_xi     = (const float*)d_in[4];
  const float* rel_o_xj = (const float*)d_in[5];
  const float* rel_o_xi = (const float*)d_in[6];
  const float* rel_p_xj = (const float*)d_in[7];
  const float* rel_p_xi = (const float*)d_in[8];
  const float* gamma    = (const float*)d_in[9];
  const float* beta     = (const float*)d_in[10];

  float* ws = (float*)d_ws;
  float* stats = ws;                  // 128 floats
  float* contrib = ws + 256;          // 64*8*1024 floats

  sp_stats<<<kB, 256, 0, stream>>>(x, stats);
  sp_main<<<kB * kD, 256, 0, stream>>>(x, o_xj, o_xi, p_xj, p_xi, rel_o_xj,
                                       rel_o_xi, rel_p_xj, rel_p_xi, gamma,
                                       beta, stats, contrib);
  sp_finish<<<(kB * kP * kP + 255) / 256, 256, 0, stream>>>(contrib,
                                                            (float*)d_out);
}